// ImageFeatureEncoder_62328565399966
// MI455X (gfx1250) — compile-verified
//
#include <hip/hip_runtime.h>

// ---------------------------------------------------------------------------
// Fully fused 3-layer GAT encoder for MI455X (gfx1250, wave32, WMMA).
//
// Roofline: 1.16 TFLOP f32 vs ~0.44 GB of HBM traffic when fully fused
// -> memory bound (~19us @ 23.3 TB/s); f32 WMMA 16x16x4 keeps reference
// precision with the compute ceiling far above the memory bound.
//
// Each 256-thread block owns 16 rows (= 4 samples of 4 graph nodes); the
// whole network for those rows is computed with the x-tile resident in LDS.
// Streaming activation I/O uses non-temporal hints so the ~34 MB weight set
// stays resident in the 192 MB L2 across all 4096 blocks.
// ---------------------------------------------------------------------------

#define NROWS   16
#define HDIM    768
#define H2DIM   1536
#define NHEADS  4
#define DOUT    192
#define NLAYERS 3
#define PITCH   772   // 768 + 4 pad: row stride 3088B -> banks rotate by 4/row
#define HPITCH  132   // 128 + 4 pad for the FFN hidden chunk buffer
#define NEG_SLOPE 0.2f
#define LN_EPS    1e-5f

typedef float v2f __attribute__((ext_vector_type(2)));
typedef float v8f __attribute__((ext_vector_type(8)));

__device__ __forceinline__ v8f wmma_f32_k4(v2f a, v2f b, v8f c) {
  // D = A(16x4) * B(4x16) + C(16x16), full f32
  return __builtin_amdgcn_wmma_f32_16x16x4_f32(
      /*neg_a=*/false, a, /*neg_b=*/false, b,
      /*c_mod=*/(short)0, c, /*reuse_a=*/false, /*reuse_b=*/false);
}

__global__ __launch_bounds__(256)
void gat_encoder_fused(const float* __restrict__ img,
                       const float* __restrict__ type_emb,
                       const float* __restrict__ Wg_all,
                       const float* __restrict__ asrc_all,
                       const float* __restrict__ adst_all,
                       const float* __restrict__ gbias_all,
                       const float* __restrict__ gamma_all,
                       const float* __restrict__ beta_all,
                       const float* __restrict__ W1_all,
                       const float* __restrict__ b1_all,
                       const float* __restrict__ W2_all,
                       const float* __restrict__ b2_all,
                       float* __restrict__ out) {
  __shared__ __align__(16) float xs[NROWS * PITCH];   // current activations x
  __shared__ __align__(16) float hs[NROWS * PITCH];   // GAT h / FFN pre-LN y
  __shared__ __align__(16) float hb[NROWS * HPITCH];  // FFN hidden chunk
  __shared__ float s_as[NROWS * NHEADS];
  __shared__ float s_ad[NROWS * NHEADS];
  __shared__ float s_alpha[NROWS * NHEADS * 4];       // [row][head][j]

  const int tid  = threadIdx.x;
  const int lane = tid & 31;
  const int wave = tid >> 5;
  const long rowbase = (long)blockIdx.x * NROWS;

  // WMMA fragment coordinates (ISA 7.12.2 layouts)
  const int arow = lane & 15;          // A: matrix row / B,C,D: column-in-tile
  const int koff = (lane >> 4) << 1;   // lanes 16..31 hold K+2,K+3
  const int rbase = (lane >> 4) << 3;  // C/D: lanes 16..31 hold M=8..15

  // ---- x = image_features + type_emb (streaming read: non-temporal) ----
  {
    int r  = tid >> 4;
    int c0 = tid & 15;
    const float* src = img + (rowbase + r) * HDIM;
    const float* te  = type_emb + (int)((rowbase + r) & 3) * HDIM;
    for (int cc = 0; cc < 48; ++cc) {
      int c = c0 + (cc << 4);
      xs[r * PITCH + c] = __builtin_nontemporal_load(&src[c]) + te[c];
    }
  }
  __syncthreads();

  for (int l = 0; l < NLAYERS; ++l) {
    const float* Wg    = Wg_all   + (size_t)l * HDIM * HDIM;
    const float* W1    = W1_all   + (size_t)l * HDIM * H2DIM;
    const float* W2    = W2_all   + (size_t)l * H2DIM * HDIM;
    const float* gamma = gamma_all + l * HDIM;
    const float* beta  = beta_all  + l * HDIM;
    const float* b1    = b1_all    + l * H2DIM;
    const float* b2    = b2_all    + l * HDIM;
    const float* gbias = gbias_all + l * HDIM;
    const float* asrc  = asrc_all  + l * NHEADS * DOUT;
    const float* adst  = adst_all  + l * NHEADS * DOUT;

    // ---------------- GEMM1: hs = xs @ Wg  (16x768x768) ----------------
    {
      v8f acc[6] = {};
      for (int kk = 0; kk < HDIM; kk += 4) {
        v2f a = *(const v2f*)&xs[arow * PITCH + kk + koff];
        const float* wrow = Wg + (size_t)(kk + koff) * HDIM;
#pragma unroll
        for (int i = 0; i < 6; ++i) {
          int col = ((wave + (i << 3)) << 4) + arow;   // tile t = wave + 8*i
          v2f b;
          b.x = wrow[col];
          b.y = wrow[HDIM + col];
          acc[i] = wmma_f32_k4(a, b, acc[i]);
        }
      }
#pragma unroll
      for (int i = 0; i < 6; ++i) {
        int col = ((wave + (i << 3)) << 4) + arow;
#pragma unroll
        for (int r = 0; r < 8; ++r)
          hs[(rbase + r) * PITCH + col] = acc[i][r];
      }
    }
    __syncthreads();

    // ---------------- attention scores (64 (row,head) pairs) ----------------
    if (tid < NROWS * NHEADS) {
      int r = tid >> 2, hh = tid & 3;
      const float* hrow = &hs[r * PITCH + hh * DOUT];
      const float* a1 = asrc + hh * DOUT;
      const float* a2 = adst + hh * DOUT;
      float ss = 0.f, sd = 0.f;
      for (int d = 0; d < DOUT; ++d) {
        float v = hrow[d];
        ss += v * a1[d];
        sd += v * a2[d];
      }
      s_as[tid] = ss;
      s_ad[tid] = sd;
    }
    __syncthreads();

    // softmax over the 4 neighbors (leaky_relu 0.2)
    if (tid < NROWS * NHEADS) {
      int r = tid >> 2, hh = tid & 3, s = r >> 2;
      float ad = s_ad[tid];
      float e[4], m = -1e30f;
#pragma unroll
      for (int j = 0; j < 4; ++j) {
        float v = ad + s_as[((s << 2) + j) * NHEADS + hh];
        v = v > 0.f ? v : NEG_SLOPE * v;
        e[j] = v;
        m = fmaxf(m, v);
      }
      float sum = 0.f;
#pragma unroll
      for (int j = 0; j < 4; ++j) { e[j] = __expf(e[j] - m); sum += e[j]; }
      float rs = 1.f / sum;
#pragma unroll
      for (int j = 0; j < 4; ++j) s_alpha[(tid << 2) + j] = e[j] * rs;
    }
    __syncthreads();

    // ---------------- aggregate + gat_bias + residual (into xs) -------------
    {
      int r  = tid >> 4;
      int c0 = tid & 15;
      int s4 = (r >> 2) << 2;
      for (int cc = 0; cc < 48; ++cc) {
        int c  = c0 + (cc << 4);
        int hh = c / DOUT;
        const float* al = &s_alpha[(r * NHEADS + hh) << 2];
        float o = al[0] * hs[(s4 + 0) * PITCH + c]
                + al[1] * hs[(s4 + 1) * PITCH + c]
                + al[2] * hs[(s4 + 2) * PITCH + c]
                + al[3] * hs[(s4 + 3) * PITCH + c];
        xs[r * PITCH + c] = o + gbias[c] + xs[r * PITCH + c];
      }
    }
    __syncthreads();

    // ---------------- LayerNorm #1 (in place on xs), wave32 reductions ------
    for (int r = wave; r < NROWS; r += 8) {
      float* rp = &xs[r * PITCH];
      float ssum = 0.f, ssq = 0.f;
      for (int c = lane; c < HDIM; c += 32) { float v = rp[c]; ssum += v; ssq += v * v; }
      for (int o = 16; o > 0; o >>= 1) {
        ssum += __shfl_xor(ssum, o, 32);
        ssq  += __shfl_xor(ssq,  o, 32);
      }
      float mn = ssum * (1.f / HDIM);
      float rv = rsqrtf(ssq * (1.f / HDIM) - mn * mn + LN_EPS);
      for (int c = lane; c < HDIM; c += 32)
        rp[c] = (rp[c] - mn) * rv * gamma[c] + beta[c];
    }
    __syncthreads();

    // ---------------- FFN: acc = relu(xs@W1+b1)@W2, hidden chunked by 128 ---
    {
      v8f acc[6] = {};
      for (int h0 = 0; h0 < H2DIM; h0 += 128) {
        // hidden[16 x 128]: one 16-col slice per wave
        v8f hacc = {};
        int hcol = (wave << 4) + arow;
        for (int kk = 0; kk < HDIM; kk += 4) {
          v2f a = *(const v2f*)&xs[arow * PITCH + kk + koff];
          const float* w0 = W1 + (size_t)(kk + koff) * H2DIM + h0 + hcol;
          v2f b;
          b.x = w0[0];
          b.y = w0[H2DIM];
          hacc = wmma_f32_k4(a, b, hacc);
        }
        float bb = b1[h0 + hcol];
#pragma unroll
        for (int r = 0; r < 8; ++r) {
          float v = hacc[r] + bb;
          hb[(rbase + r) * HPITCH + hcol] = v > 0.f ? v : 0.f;  // relu
        }
        __syncthreads();

        // acc += hidden_chunk @ W2[h0:h0+128, :]
        for (int kk = 0; kk < 128; kk += 4) {
          v2f a = *(const v2f*)&hb[arow * HPITCH + kk + koff];
          const float* w0 = W2 + (size_t)(h0 + kk + koff) * HDIM;
#pragma unroll
          for (int i = 0; i < 6; ++i) {
            int col = ((wave + (i << 3)) << 4) + arow;
            v2f b;
            b.x = w0[col];
            b.y = w0[HDIM + col];
            acc[i] = wmma_f32_k4(a, b, acc[i]);
          }
        }
        __syncthreads();
      }
      // epilogue: hs = acc + b2 + xs (pre-LN y)
#pragma unroll
      for (int i = 0; i < 6; ++i) {
        int col = ((wave + (i << 3)) << 4) + arow;
        float bb = b2[col];
#pragma unroll
        for (int r = 0; r < 8; ++r) {
          int rr = rbase + r;
          hs[rr * PITCH + col] = acc[i][r] + bb + xs[rr * PITCH + col];
        }
      }
    }
    __syncthreads();

    // ---------------- LayerNorm #2 (same gamma/beta): xs = LN(hs) -----------
    for (int r = wave; r < NROWS; r += 8) {
      const float* rp = &hs[r * PITCH];
      float* op = &xs[r * PITCH];
      float ssum = 0.f, ssq = 0.f;
      for (int c = lane; c < HDIM; c += 32) { float v = rp[c]; ssum += v; ssq += v * v; }
      for (int o = 16; o > 0; o >>= 1) {
        ssum += __shfl_xor(ssum, o, 32);
        ssq  += __shfl_xor(ssq,  o, 32);
      }
      float mn = ssum * (1.f / HDIM);
      float rv = rsqrtf(ssq * (1.f / HDIM) - mn * mn + LN_EPS);
      for (int c = lane; c < HDIM; c += 32)
        op[c] = (rp[c] - mn) * rv * gamma[c] + beta[c];
    }
    __syncthreads();
  }

  // ---- store final x (streaming write: non-temporal) ----
  {
    int r  = tid >> 4;
    int c0 = tid & 15;
    float* dst = out + (rowbase + r) * HDIM;
    for (int cc = 0; cc < 48; ++cc) {
      int c = c0 + (cc << 4);
      __builtin_nontemporal_store(xs[r * PITCH + c], &dst[c]);
    }
  }
}

extern "C" void kernel_launch(void* const* d_in, const int* in_sizes, int n_in,
                              void* d_out, int out_size, void* d_ws, size_t ws_size,
                              hipStream_t stream) {
  (void)in_sizes; (void)n_in; (void)d_ws; (void)ws_size; (void)out_size;
  const float* img   = (const float*)d_in[0];   // [16384,4,768]
  const float* temb  = (const float*)d_in[1];   // [4,768]
  const float* Wg    = (const float*)d_in[2];   // [3,768,768]
  const float* asrc  = (const float*)d_in[3];   // [3,4,192]
  const float* adst  = (const float*)d_in[4];   // [3,4,192]
  const float* gbias = (const float*)d_in[5];   // [3,768]
  const float* gamma = (const float*)d_in[6];   // [3,768]
  const float* beta  = (const float*)d_in[7];   // [3,768]
  const float* W1    = (const float*)d_in[8];   // [3,768,1536]
  const float* b1    = (const float*)d_in[9];   // [3,1536]
  const float* W2    = (const float*)d_in[10];  // [3,1536,768]
  const float* b2    = (const float*)d_in[11];  // [3,768]
  float* out = (float*)d_out;                   // [16384,4,768]

  const int total_rows = 16384 * 4;
  dim3 grid(total_rows / NROWS), block(256);
  gat_encoder_fused<<<grid, block, 0, stream>>>(
      img, temb, Wg, asrc, adst, gbias, gamma, beta, W1, b1, W2, b2, out);
}